// HyperGANUpBlock_49795850830343
// MI455X (gfx1250) — compile-verified
//
#include <hip/hip_runtime.h>
#include <hip/hip_bf16.h>

// ---------------------------------------------------------------------------
// HyperGAN up-block for MI455X (gfx1250, wave32, WMMA).
// upsample(2x) -> per-sample conv3d(3^3, 256->128) -> IN+relu -> concat(skip)
// -> per-sample conv3d(3^3, 256->128) -> IN+relu
//
// Implicit-GEMM conv using V_WMMA_F32_16X16X4_F32 (fp32 fidelity; workload is
// memory-bound at 23.3 TB/s so low precision buys nothing). LDS tiles are
// stored in fragment-native layouts so A/B WMMA operands are single aligned
// ds_load_b64s, and the K-loop is double-buffered: next chunk's global gather
// issues before the current chunk's WMMA stream (one barrier per phase).
// ---------------------------------------------------------------------------

typedef float v2f __attribute__((ext_vector_type(2)));
typedef float v8f __attribute__((ext_vector_type(8)));

#define BATCH   4
#define ODIM    32                       // output spatial extent per axis
#define NVOX    (ODIM*ODIM*ODIM)         // 32768 voxels per sample
#define COUT    128
#define CINTOT  256                      // conv1: 256 cin; conv2: 128 y1 + 128 skip
#define KC      32                       // cin chunk per LDS phase
#define MT      64                       // voxel (M) tile per workgroup
#define NPHASE  (27 * (CINTOT / KC))     // 216 K-phases
#define EPSV    1e-5f

#define APAD    (KC + 2)                 // 34: even (b64-aligned) + conflict-free
#define BSZ     ((KC/4) * COUT * 4)      // B frag layout: [k/4][n][k%4]

// Workspace layout (floats)
#define WS_Y1     0                          // [4][32768][128] raw conv1 out
#define WS_PART   (BATCH*NVOX*COUT)          // [4][64][128][2] partial sums
#define WS_STAT1  (WS_PART + BATCH*64*COUT*2)
#define WS_STAT2  (WS_STAT1 + BATCH*COUT*2)

// ---------------------------------------------------------------------------
// Stage one K-phase: B tile (weights, contiguous slab -> fragment layout) and
// A tile (stencil gather; pass1 folds nearest-up2, pass2 folds IN+relu+concat).
// ---------------------------------------------------------------------------
template <int PASS>
__device__ __forceinline__ void stage_tiles(
    float* __restrict__ As, float* __restrict__ Bs,
    const float* __restrict__ src, const float* __restrict__ skip,
    const float* __restrict__ w,
    const float* __restrict__ s_mean, const float* __restrict__ s_rstd,
    int b, int m0, int ph, int tid)
{
    const int tap = ph >> 3;
    const int k0  = (ph & 7) * KC;
    const int dz = tap / 9 - 1;
    const int dy = (tap / 3) % 3 - 1;
    const int dx = tap % 3 - 1;

    // ---- B: w[b, tap, k0:k0+KC, 0:128] -> Bs[(k>>2)*512 + n*4 + (k&3)]
    const float* wb = w + (((size_t)b * 27 + tap) * CINTOT + k0) * COUT;
    #pragma unroll
    for (int i = 0; i < (KC * COUT) / 256; ++i) {
        int el = tid + i * 256;
        int k = el >> 7, n = el & 127;
        Bs[(k >> 2) * (COUT * 4) + n * 4 + (k & 3)] = wb[el];
    }

    // ---- A: 64 voxels x KC channels
    #pragma unroll
    for (int i = 0; i < (MT * KC) / 256; ++i) {
        int el   = tid + i * 256;
        int vox  = el >> 5;
        int cidx = el & (KC - 1);
        int cin  = k0 + cidx;
        int p    = m0 + vox;
        int z = p >> 10, y = (p >> 5) & 31, x = p & 31;
        int zz = z + dz, yy = y + dy, xx = x + dx;
        bool inb = (unsigned)zz < (unsigned)ODIM &&
                   (unsigned)yy < (unsigned)ODIM &&
                   (unsigned)xx < (unsigned)ODIM;
        float v = 0.f;
        if (PASS == 1) {
            if (inb) {      // nearest-neighbor upsample: halve source coords
                int si = (((b * 16 + (zz >> 1)) * 16 + (yy >> 1)) * 16
                          + (xx >> 1)) * 256 + cin;
                v = src[si];
            }
        } else {
            if (inb) {
                int sp = (zz * ODIM + yy) * ODIM + xx;
                if (cin < COUT) {
                    float u = src[((size_t)b * NVOX + sp) * COUT + cin];
                    v = fmaxf(0.f, (u - s_mean[cin]) * s_rstd[cin]);
                } else {
                    v = skip[((size_t)b * NVOX + sp) * COUT + (cin - COUT)];
                }
            }
        }
        As[vox * APAD + cidx] = v;
    }
}

// ---------------------------------------------------------------------------
// 8 WMMA K-steps over one staged chunk. A frag: (K=kk+2*half, kk+2*half+1) at
// even LDS index -> one ds_load_b64. B frag likewise via fragment layout.
// ---------------------------------------------------------------------------
__device__ __forceinline__ void compute_tiles(
    const float* __restrict__ As, const float* __restrict__ Bs,
    v8f acc[4], int wm, int wn, int half, int l16)
{
    const int arow = wm * 16 + l16;
    #pragma unroll
    for (int kk = 0; kk < KC; kk += 4) {
        v2f a = *(const v2f*)(As + arow * APAD + kk + 2 * half);
        #pragma unroll
        for (int t = 0; t < 4; ++t) {
            int ncol = wn * 64 + t * 16 + l16;
            v2f bf = *(const v2f*)(Bs + (kk >> 2) * (COUT * 4) + ncol * 4 + 2 * half);
            acc[t] = __builtin_amdgcn_wmma_f32_16x16x4_f32(
                         false, a, false, bf, (short)0, acc[t], false, false);
        }
    }
}

// ---------------------------------------------------------------------------
// Implicit-GEMM conv3d, SAME padding, per-sample weights.
// grid = BATCH * (NVOX/MT) blocks, 256 threads (8 waves).
// Wave (wm, wn) owns a 16(M) x 64(N) strip = 4 WMMA C tiles.
// ---------------------------------------------------------------------------
template <int PASS>
__global__ __launch_bounds__(256)
void conv3d_wmma(const float* __restrict__ src,
                 const float* __restrict__ skip,
                 const float* __restrict__ w,      // [4,3,3,3,CINTOT,128]
                 const float* __restrict__ stats,  // pass2 only
                 float* __restrict__ dst)          // [4,32768,128] raw
{
    __shared__ __align__(16) float As[2][MT * APAD];
    __shared__ __align__(16) float Bs[2][BSZ];
    __shared__ float s_mean[COUT];
    __shared__ float s_rstd[COUT];

    const int tid = threadIdx.x;
    const int b   = blockIdx.x / (NVOX / MT);
    const int m0  = (blockIdx.x % (NVOX / MT)) * MT;

    if (PASS == 2) {
        if (tid < COUT) {
            s_mean[tid] = stats[(b * COUT + tid) * 2 + 0];
            s_rstd[tid] = stats[(b * COUT + tid) * 2 + 1];
        }
        __syncthreads();
    }

    const int wave = tid >> 5;
    const int lane = tid & 31;
    const int wm   = wave & 3;          // M sub-tile: 4 x 16 voxels
    const int wn   = wave >> 2;         // N half:     2 x 64 channels
    const int half = lane >> 4;
    const int l16  = lane & 15;

    v8f acc[4] = {{}, {}, {}, {}};

    // Double-buffered pipeline: stage(ph+1) overlaps compute(ph).
    stage_tiles<PASS>(As[0], Bs[0], src, skip, w, s_mean, s_rstd, b, m0, 0, tid);
    __syncthreads();

    int cb = 0;
    for (int ph = 0; ph < NPHASE; ++ph) {
        if (ph + 1 < NPHASE)
            stage_tiles<PASS>(As[cb ^ 1], Bs[cb ^ 1], src, skip, w,
                              s_mean, s_rstd, b, m0, ph + 1, tid);
        compute_tiles(As[cb], Bs[cb], acc, wm, wn, half, l16);
        __syncthreads();
        cb ^= 1;
    }

    // ---- epilogue: C layout VGPR r -> M = r + 8*half, N = l16
    #pragma unroll
    for (int t = 0; t < 4; ++t) {
        int co = wn * 64 + t * 16 + l16;
        float* o = dst + ((size_t)b * NVOX + m0 + wm * 16) * COUT + co;
        #pragma unroll
        for (int r = 0; r < 8; ++r) {
            int M = r + half * 8;
            o[(size_t)M * COUT] = acc[t][r];
        }
    }
}

// ---------------------------------------------------------------------------
// Deterministic two-stage instance-norm statistics (bitwise reproducible
// across graph replays; no float atomics).
// Stage 1: grid = BATCH*64; block reduces 512 voxels, thread t owns channel
// (t & 127), fully coalesced.
// ---------------------------------------------------------------------------
__global__ __launch_bounds__(256)
void in_stats_partial(const float* __restrict__ in, float* __restrict__ part)
{
    __shared__ float ls[256], lss[256];
    const int tid = threadIdx.x;
    const int b   = blockIdx.x >> 6;
    const int ch  = blockIdx.x & 63;
    const float* p = in + (size_t)b * NVOX * COUT + (size_t)ch * 512 * COUT;

    float s = 0.f, ss = 0.f;
    #pragma unroll 4
    for (int i = 0; i < (512 * COUT) / 256; ++i) {
        float v = p[tid + i * 256];
        s += v; ss += v * v;
    }
    ls[tid] = s; lss[tid] = ss;
    __syncthreads();
    if (tid < COUT) {
        float s2  = ls[tid]  + ls[tid + COUT];
        float ss2 = lss[tid] + lss[tid + COUT];
        float* o = part + ((size_t)(b * 64 + ch) * COUT + tid) * 2;
        o[0] = s2; o[1] = ss2;
    }
}

// Stage 2: one block, 512 threads -> (b,c) mean & rstd.
__global__ __launch_bounds__(512)
void in_stats_final(const float* __restrict__ part, float* __restrict__ stats)
{
    const int t = threadIdx.x;            // t = b*128 + c
    const int b = t >> 7, c = t & 127;
    float s = 0.f, ss = 0.f;
    for (int ch = 0; ch < 64; ++ch) {
        const float* p = part + ((size_t)(b * 64 + ch) * COUT + c) * 2;
        s += p[0]; ss += p[1];
    }
    const float inv_n = 1.0f / (float)NVOX;
    float mean = s * inv_n;
    float var  = fmaxf(0.f, ss * inv_n - mean * mean);
    stats[t * 2 + 0] = mean;
    stats[t * 2 + 1] = __frsqrt_rn(var + EPSV);
}

// Final elementwise normalize + relu over d_out. NVOX*COUT = 2^22 per sample.
__global__ __launch_bounds__(256)
void in_norm_relu(float* __restrict__ out, const float* __restrict__ stats)
{
    __shared__ float sm[BATCH * COUT], sr[BATCH * COUT];
    const int tid = threadIdx.x;
    #pragma unroll
    for (int i = 0; i < (BATCH * COUT) / 256; ++i) {
        sm[tid + i * 256] = stats[(tid + i * 256) * 2 + 0];
        sr[tid + i * 256] = stats[(tid + i * 256) * 2 + 1];
    }
    __syncthreads();

    const size_t total  = (size_t)BATCH * NVOX * COUT;
    const size_t stride = (size_t)gridDim.x * blockDim.x;
    for (size_t i = (size_t)blockIdx.x * blockDim.x + tid; i < total; i += stride) {
        int b = (int)(i >> 22);           // NVOX*COUT == 4194304 == 2^22
        int c = (int)(i & 127);
        int sc = b * COUT + c;
        out[i] = fmaxf(0.f, (out[i] - sm[sc]) * sr[sc]);
    }
}

// ---------------------------------------------------------------------------
// Inputs (setup_inputs order): x, w1, w2, skip, training
// ---------------------------------------------------------------------------
extern "C" void kernel_launch(void* const* d_in, const int* in_sizes, int n_in,
                              void* d_out, int out_size, void* d_ws, size_t ws_size,
                              hipStream_t stream) {
    const float* x    = (const float*)d_in[0];
    const float* w1   = (const float*)d_in[1];
    const float* w2   = (const float*)d_in[2];
    const float* skip = (const float*)d_in[3];
    float* out = (float*)d_out;
    float* ws  = (float*)d_ws;

    float* y1     = ws + WS_Y1;
    float* part   = ws + WS_PART;
    float* stats1 = ws + WS_STAT1;
    float* stats2 = ws + WS_STAT2;

    const dim3 blk(256);
    const dim3 conv_grid(BATCH * (NVOX / MT));   // 2048 workgroups

    // conv1 (upsample fused) -> y1 raw
    conv3d_wmma<1><<<conv_grid, blk, 0, stream>>>(x, nullptr, w1, nullptr, y1);
    // instance-norm stats of y1
    in_stats_partial<<<dim3(BATCH * 64), blk, 0, stream>>>(y1, part);
    in_stats_final<<<dim3(1), dim3(512), 0, stream>>>(part, stats1);
    // conv2 (norm+relu+concat fused into gather) -> out raw
    conv3d_wmma<2><<<conv_grid, blk, 0, stream>>>(y1, skip, w2, stats1, out);
    // instance-norm stats of out, then normalize+relu in place
    in_stats_partial<<<dim3(BATCH * 64), blk, 0, stream>>>(out, part);
    in_stats_final<<<dim3(1), dim3(512), 0, stream>>>(part, stats2);
    in_norm_relu<<<dim3(2048), blk, 0, stream>>>(out, stats2);
}